// MultiHeadAttentionRelative_88862873354524
// MI455X (gfx1250) — compile-verified
//
#include <hip/hip_runtime.h>

#define DIMS 128
#define NHEADS 4
#define JLEN 81
#define JT 16
#define NT 6              // 6*16 = 96 >= 81 j-slots
#define ROWF 132          // padded LDS row stride in floats (528B: 16B aligned, bank-spread)
#define TILEF (JT * ROWF) // 2112 floats per tile
#define WPB 2             // waves (batches) per block
#define WAVE_F (512 + 2 * 3 * TILEF) // q(128)+scores(384) + 2 buf * 3 arrays
#define SCALE_F 0.17677669529663687f // 32^-0.5

typedef float v2f __attribute__((ext_vector_type(2)));
typedef float v8f __attribute__((ext_vector_type(8)));
typedef int v4i __attribute__((vector_size(4 * sizeof(int))));
typedef __attribute__((address_space(1))) v4i g_v4i; // global (printed as __device__)
typedef __attribute__((address_space(3))) v4i l_v4i; // LDS

#if defined(__has_builtin)
#if __has_builtin(__builtin_amdgcn_global_load_async_to_lds_b128)
#define HAVE_ASYNC_BUILTIN 1
#endif
#if __has_builtin(__builtin_amdgcn_s_wait_asynccnt)
#define HAVE_WAITA_BUILTIN 1
#endif
#endif

__device__ __forceinline__ void async_copy_b128(const void* g, void* l) {
#ifdef HAVE_ASYNC_BUILTIN
  __builtin_amdgcn_global_load_async_to_lds_b128((g_v4i*)g, (l_v4i*)l, 0, 0);
#else
  asm volatile("global_load_async_to_lds_b128 %0, %1, off"
               :: "v"((l_v4i*)l), "v"((g_v4i*)g) : "memory");
#endif
}
__device__ __forceinline__ void wait_async_le(int n) {
#ifdef HAVE_WAITA_BUILTIN
  if (n == 0) __builtin_amdgcn_s_wait_asynccnt(0);
  else        __builtin_amdgcn_s_wait_asynccnt(48);
#else
  if (n == 0) asm volatile("s_wait_asynccnt 0" ::: "memory");
  else        asm volatile("s_wait_asynccnt 48" ::: "memory");
#endif
  // compiler scheduling barrier: LDS reads of the arrived tile must not be
  // hoisted above the asynccnt wait (async LDS writes are invisible to AA)
  asm volatile("" ::: "memory");
}

__global__ __launch_bounds__(WPB * 32) void
mhra_kernel(const float* __restrict__ Q, const float* __restrict__ K,
            const float* __restrict__ V, const float* __restrict__ Qr,
            const float* __restrict__ Kr, float* __restrict__ out, int B) {
  __shared__ float smem[WPB * WAVE_F];
  const int lane = threadIdx.x & 31;
  const int wid  = threadIdx.x >> 5;
  const int b = blockIdx.x * WPB + wid;
  if (b >= B) return; // wave-uniform: EXEC stays all-ones below

  float* my = smem + wid * WAVE_F;
  float* qld   = my;         // 128 floats
  float* sc    = my + 128;   // 4 * 96 floats
  float* tiles = my + 512;   // 2 buffers * 3 arrays * TILEF

  const size_t bo = (size_t)b * JLEN * DIMS;
  const float* Kb  = K  + bo;
  const float* Krb = Kr + bo;
  const float* Qrb = Qr + bo;
  const float* Vb  = V  + bo;

  // stage q into LDS (lane l owns dims 4l..4l+3)
  *(float4*)(qld + lane * 4) = *(const float4*)(Q + (size_t)b * DIMS + lane * 4);

  auto issue_tile = [&](int jt) {
    float* buf = tiles + (jt & 1) * (3 * TILEF);
    const float* srcs[3] = {Kb, Krb, Qrb};
    for (int a = 0; a < 3; ++a) {
      const float* s = srcs[a];
      float* t = buf + a * TILEF;
      for (int r = 0; r < JT; ++r) {
        int jj = jt * JT + r;
        if (jj > JLEN - 1) jj = JLEN - 1; // clamp padding rows (never read)
        async_copy_b128((const char*)(s + (size_t)jj * DIMS) + lane * 16,
                        (char*)(t + r * ROWF) + lane * 16);
      }
    }
  };

  issue_tile(0);
  const int hi2 = (lane >= 16) ? 2 : 0; // K-slot offset for upper half-wave
  const int m   = lane & 15;            // WMMA row (A) / column j (B)
  const v2f vz  = {0.0f, 0.0f};
  const v2f v1  = {1.0f, 1.0f};

  for (int jt = 0; jt < NT; ++jt) {
    if (jt + 1 < NT) issue_tile(jt + 1);
    wait_async_le((jt + 1 < NT) ? 48 : 0); // current tile resident in LDS

    const float* buf = tiles + (jt & 1) * (3 * TILEF);
    const float* tK  = buf;
    const float* tKr = buf + TILEF;
    const float* tQr = buf + 2 * TILEF;
    const int rb = m * ROWF;

    v8f acc = {0.f, 0.f, 0.f, 0.f, 0.f, 0.f, 0.f, 0.f};
#pragma unroll 4
    for (int d0 = 0; d0 < DIMS; d0 += 4) {
      const bool sel = (m == (d0 >> 5)); // row m active only in head-m band
      v2f aq = *(const v2f*)(qld + d0 + hi2);
      aq = sel ? aq : vz;                // A for c_c / c_p: band-masked q
      const v2f a1 = sel ? v1 : vz;      // A for p_c: band-masked ones
      const v2f bK  = *(const v2f*)(tK  + rb + d0 + hi2);
      const v2f bKr = *(const v2f*)(tKr + rb + d0 + hi2);
      const v2f bQr = *(const v2f*)(tQr + rb + d0 + hi2);
      const v2f bP  = bK * bQr;          // rowwise Qr*K feeds p_c
      acc = __builtin_amdgcn_wmma_f32_16x16x4_f32(false, aq, false, bK,  (short)0, acc, false, false);
      acc = __builtin_amdgcn_wmma_f32_16x16x4_f32(false, aq, false, bKr, (short)0, acc, false, false);
      acc = __builtin_amdgcn_wmma_f32_16x16x4_f32(false, a1, false, bP,  (short)0, acc, false, false);
    }
    // rows 0..3 of D are the 4 heads' scores for j = jt*16 + (lane%16)
    if (lane < 16) {
      const int j = jt * JT + lane;
      if (j < JLEN) {
        sc[0 * 96 + j] = acc[0] * SCALE_F;
        sc[1 * 96 + j] = acc[1] * SCALE_F;
        sc[2 * 96 + j] = acc[2] * SCALE_F;
        sc[3 * 96 + j] = acc[3] * SCALE_F;
      }
    }
  }

  // softmax per head: 8-lane group h = lane>>3 owns head h
  const int h = lane >> 3, r = lane & 7;
  float mx = -3.4e38f;
  for (int j = r; j < JLEN; j += 8) mx = fmaxf(mx, sc[h * 96 + j]);
  mx = fmaxf(mx, __shfl_xor(mx, 4, 8));
  mx = fmaxf(mx, __shfl_xor(mx, 2, 8));
  mx = fmaxf(mx, __shfl_xor(mx, 1, 8));
  float sum = 0.f;
  for (int j = r; j < JLEN; j += 8) {
    const float e = __expf(sc[h * 96 + j] - mx);
    sc[h * 96 + j] = e;
    sum += e;
  }
  sum += __shfl_xor(sum, 4, 8);
  sum += __shfl_xor(sum, 2, 8);
  sum += __shfl_xor(sum, 1, 8);
  const float inv = 1.0f / sum;

  // out[h*32 + (r*4..r*4+3)] = inv * sum_j e[h,j] * V[j, 4*lane..]
  float4 o = {0.f, 0.f, 0.f, 0.f};
  for (int j = 0; j < JLEN; ++j) {
    const float w = sc[h * 96 + j];
    const float4 v4 = *(const float4*)(Vb + (size_t)j * DIMS + lane * 4);
    o.x += w * v4.x; o.y += w * v4.y; o.z += w * v4.z; o.w += w * v4.w;
  }
  o.x *= inv; o.y *= inv; o.z *= inv; o.w *= inv;
  *(float4*)(out + (size_t)b * DIMS + lane * 4) = o;
}

extern "C" void kernel_launch(void* const* d_in, const int* in_sizes, int n_in,
                              void* d_out, int out_size, void* d_ws, size_t ws_size,
                              hipStream_t stream) {
  (void)n_in; (void)out_size; (void)d_ws; (void)ws_size;
  const float* Q  = (const float*)d_in[0];
  const float* K  = (const float*)d_in[1];
  const float* V  = (const float*)d_in[2];
  const float* Qr = (const float*)d_in[3];
  const float* Kr = (const float*)d_in[4];
  float* out = (float*)d_out;
  const int B = in_sizes[0] / DIMS;
  const int blocks = (B + WPB - 1) / WPB;
  hipLaunchKernelGGL(mhra_kernel, dim3(blocks), dim3(WPB * 32), 0, stream,
                     Q, K, V, Qr, Kr, out, B);
}